// GGNNMeanEnd2End_3298534883491
// MI455X (gfx1250) — compile-verified
//
#include <hip/hip_runtime.h>
#include <hip/hip_bf16.h>

#define N_NODES  50000
#define N_EDGES  800000
#define DIM      128
#define N_ETYPES 4
#define N_STEPS  8
#define B_GRAPHS 50
#define HID      256
#define NKEYS    (N_NODES * N_ETYPES)   // 200000 (node,etype) buckets

// LDS row strides chosen for bank-conflict-free fragment reads/writes
#define XPAD 132     // activation slab row stride
#define GPAD 388     // gi/gh row stride

typedef __attribute__((ext_vector_type(2))) float v2f;
typedef __attribute__((ext_vector_type(8))) float v8f;

// ---------------------------------------------------------------------------
// One-shot weight repacks into fragment-contiguous float2 layouts.
// msg:  pk[((t*32 + k4)*2 + half)*128 + col] = {W[t][4k4+2half][col], W[t][4k4+2half+1][col]}
// gru:  pk[(k4*2 + half)*384 + c]            = {W[c][4k4+2half],     W[c][4k4+2half+1]}
// ---------------------------------------------------------------------------
__global__ void pack_msg_kernel(const float* __restrict__ Wm, float2* __restrict__ pk) {
    int idx = blockIdx.x * 256 + threadIdx.x;           // 4*32*2*128 = 32768
    if (idx >= N_ETYPES * 32 * 2 * DIM) return;
    int col  = idx & 127;
    int half = (idx >> 7) & 1;
    int k4   = (idx >> 8) & 31;
    int t    = idx >> 13;
    int k    = k4 * 4 + half * 2;
    pk[idx] = make_float2(Wm[(size_t)t * DIM * DIM + (size_t)k * DIM + col],
                          Wm[(size_t)t * DIM * DIM + (size_t)(k + 1) * DIM + col]);
}

__global__ void pack_gru_kernel(const float* __restrict__ W, float2* __restrict__ pk) {
    int idx = blockIdx.x * 256 + threadIdx.x;           // 32*2*384 = 24576
    if (idx >= 32 * 2 * 384) return;
    int c    = idx % 384;
    int half = (idx / 384) & 1;
    int k4   = idx / 768;
    int k    = k4 * 4 + half * 2;
    pk[idx] = make_float2(W[(size_t)c * DIM + k], W[(size_t)c * DIM + k + 1]);
}

// ---------------------------------------------------------------------------
// CSR build (one-time; edge structure is static across steps)
// ---------------------------------------------------------------------------
__global__ void count_edges_kernel(const int* __restrict__ dst,
                                   const int* __restrict__ et,
                                   int* __restrict__ cnt) {
    int e = blockIdx.x * blockDim.x + threadIdx.x;
    if (e < N_EDGES) atomicAdd(&cnt[dst[e] * N_ETYPES + et[e]], 1);
}

// two-level exclusive scan: 1024 elements / block (256 thr x 4)
__global__ void scan1_kernel(const int* __restrict__ in, int* __restrict__ out,
                             int* __restrict__ bsum, int M) {
    __shared__ int tsum[256];
    int tid  = threadIdx.x;
    int base = blockIdx.x * 1024 + tid * 4;
    int v0 = (base + 0 < M) ? in[base + 0] : 0;
    int v1 = (base + 1 < M) ? in[base + 1] : 0;
    int v2 = (base + 2 < M) ? in[base + 2] : 0;
    int v3 = (base + 3 < M) ? in[base + 3] : 0;
    int s = v0 + v1 + v2 + v3;
    tsum[tid] = s;
    __syncthreads();
    for (int offs = 1; offs < 256; offs <<= 1) {
        int x = (tid >= offs) ? tsum[tid - offs] : 0;
        __syncthreads();
        tsum[tid] += x;
        __syncthreads();
    }
    int texc = tsum[tid] - s;   // exclusive prefix of this thread's chunk
    if (base + 0 < M) out[base + 0] = texc;
    if (base + 1 < M) out[base + 1] = texc + v0;
    if (base + 2 < M) out[base + 2] = texc + v0 + v1;
    if (base + 3 < M) out[base + 3] = texc + v0 + v1 + v2;
    if (tid == 255) bsum[blockIdx.x] = tsum[255];
}

__global__ void scan2_kernel(int* __restrict__ bsum, int n) {   // n <= 256
    __shared__ int tmp[256];
    int tid = threadIdx.x;
    int v = (tid < n) ? bsum[tid] : 0;
    tmp[tid] = v;
    __syncthreads();
    for (int offs = 1; offs < 256; offs <<= 1) {
        int x = (tid >= offs) ? tmp[tid - offs] : 0;
        __syncthreads();
        tmp[tid] += x;
        __syncthreads();
    }
    if (tid < n) bsum[tid] = tmp[tid] - v;  // exclusive
}

__global__ void scan3_kernel(int* __restrict__ out, const int* __restrict__ bsum, int M) {
    int i = blockIdx.x * 1024 + threadIdx.x * 4;
    int add = bsum[blockIdx.x];
    if (i + 0 < M) out[i + 0] += add;
    if (i + 1 < M) out[i + 1] += add;
    if (i + 2 < M) out[i + 2] += add;
    if (i + 3 < M) out[i + 3] += add;
}

__global__ void fill_csr_kernel(const int* __restrict__ src,
                                const int* __restrict__ dst,
                                const int* __restrict__ et,
                                int* __restrict__ cursor,
                                int* __restrict__ eidx) {
    int e = blockIdx.x * blockDim.x + threadIdx.x;
    if (e >= N_EDGES) return;
    int slot = atomicAdd(&cursor[dst[e] * N_ETYPES + et[e]], 1);
    eidx[slot] = src[e];
}

// ---------------------------------------------------------------------------
// Message GEMM with fused CSR gather-aggregation:
//   A[N,128] = sum_t ( sum_{e in bucket(v,t)} h[src_e] ) @ W_msg[t] + cnt*b_msg
// 16-row slab per block, 4 waves. Each wave gather-sums its buckets' source
// rows (coalesced float4 per lane, register accumulate, single LDS store),
// then fp32 WMMA 16x16x4 against fragment-packed weights.
// ---------------------------------------------------------------------------
__global__ void msg_gemm_kernel(const float* __restrict__ h,
                                const int* __restrict__ cnt,
                                const int* __restrict__ eoff,
                                const int* __restrict__ eidx,
                                const float2* __restrict__ pkm,
                                const float* __restrict__ bm,
                                float* __restrict__ A) {
    __shared__ float sS[N_ETYPES * 16 * XPAD];   // [t][row][XPAD]
    const int slab    = blockIdx.x;              // 16 rows
    const int lane    = threadIdx.x & 31;
    const int wave    = threadIdx.x >> 5;        // 0..3
    const int colbase = wave * 32;
    const int mrow    = lane & 15;
    const int koff    = (lane >> 4) * 2;
    const int half    = lane >> 4;
    const int ncol    = lane & 15;

    // gather-aggregate: 64 (row,type) buckets, 16 per wave
    for (int b = wave; b < 64; b += 4) {
        int r = b >> 2;
        int t = b & 3;
        int key = (slab * 16 + r) * N_ETYPES + t;
        int beg = eoff[key];
        int num = cnt[key];
        float ax = 0.f, ay = 0.f, az = 0.f, aw = 0.f;
        for (int i = 0; i < num; ++i) {
            int s = eidx[beg + i];
            const float4 v = *(const float4*)(h + (size_t)s * DIM + lane * 4);
            ax += v.x; ay += v.y; az += v.z; aw += v.w;
        }
        float* dp = &sS[(t * 16 + r) * XPAD + lane * 4];
        dp[0] = ax; dp[1] = ay; dp[2] = az; dp[3] = aw;
    }
    __syncthreads();

    v8f c0 = {}, c1 = {};
    for (int t = 0; t < N_ETYPES; ++t) {
        const float* srow = &sS[(t * 16 + mrow) * XPAD];
        #pragma unroll 4
        for (int k = 0; k < DIM; k += 4) {
            v2f a;
            a.x = srow[k + koff];
            a.y = srow[k + koff + 1];
            const float2* pb = pkm + ((size_t)((t * 32 + (k >> 2)) * 2 + half)) * DIM;
            float2 f0 = pb[colbase + ncol];
            float2 f1 = pb[colbase + 16 + ncol];
            v2f b0; b0.x = f0.x; b0.y = f0.y;
            v2f b1; b1.x = f1.x; b1.y = f1.y;
            c0 = __builtin_amdgcn_wmma_f32_16x16x4_f32(false, a, false, b0,
                                                       (short)0, c0, false, false);
            c1 = __builtin_amdgcn_wmma_f32_16x16x4_f32(false, a, false, b1,
                                                       (short)0, c1, false, false);
        }
    }
    #pragma unroll
    for (int j = 0; j < 8; ++j) {
        int r    = slab * 16 + j + half * 8;
        int col0 = colbase + ncol;
        int col1 = colbase + 16 + ncol;
        float bias0 = 0.f, bias1 = 0.f;
        #pragma unroll
        for (int t = 0; t < N_ETYPES; ++t) {
            float cf = (float)cnt[r * N_ETYPES + t];
            bias0 += cf * bm[t * DIM + col0];
            bias1 += cf * bm[t * DIM + col1];
        }
        A[(size_t)r * DIM + col0] = c0[j] + bias0;
        A[(size_t)r * DIM + col1] = c1[j] + bias1;
    }
}

// ---------------------------------------------------------------------------
// Fused GRU: gi = A @ Wih^T, gh = h @ Whh^T (fp32 WMMA), gates in LDS,
// in-place h update. 8 waves: 0-3 -> gi (6 tiles each), 4-7 -> gh.
// ---------------------------------------------------------------------------
__global__ void gru_fused_kernel(const float* __restrict__ A,
                                 float* __restrict__ h,
                                 const float2* __restrict__ pkih,
                                 const float2* __restrict__ pkhh,
                                 const float* __restrict__ bih,
                                 const float* __restrict__ bhh) {
    __shared__ float lds[2 * 16 * GPAD];         // 49664 B, overlaid usage
    float* sA = lds;                             // 16 x XPAD   (phase 1)
    float* sH = lds + 16 * XPAD;                 // 16 x XPAD   (phase 1)
    float* gi = lds;                             // 16 x GPAD   (phase 2)
    float* gh = lds + 16 * GPAD;                 // 16 x GPAD   (phase 2)

    const int slab = blockIdx.x;
    const int lane = threadIdx.x & 31;
    const int wave = threadIdx.x >> 5;           // 0..7
    const bool is_h = wave >= 4;
    const int  w4   = wave & 3;
    const int mrow  = lane & 15;
    const int koff  = (lane >> 4) * 2;
    const int half  = lane >> 4;
    const int ncol  = lane & 15;

    for (int i = threadIdx.x; i < 1024; i += 256) {
        int which = i >> 9;                      // 0 = A, 1 = h
        int rem   = i & 511;
        int r     = rem >> 5;
        int c4    = rem & 31;
        const float* src = (which ? h : A) + (size_t)(slab * 16 + r) * DIM + c4 * 4;
        float4 v = *(const float4*)src;
        float* dp = (which ? sH : sA) + r * XPAD + c4 * 4;
        dp[0] = v.x; dp[1] = v.y; dp[2] = v.z; dp[3] = v.w;
    }
    __syncthreads();

    const float* xrow = (is_h ? sH : sA) + mrow * XPAD;
    const float2* Wp  = is_h ? pkhh : pkih;
    v8f acc[6] = {};
    for (int k = 0; k < DIM; k += 4) {
        v2f a;
        a.x = xrow[k + koff];
        a.y = xrow[k + koff + 1];
        const float2* pb = Wp + (size_t)((k >> 2) * 2 + half) * 384;
        #pragma unroll
        for (int tIdx = 0; tIdx < 6; ++tIdx) {
            int col = (w4 * 6 + tIdx) * 16 + ncol;
            float2 f = pb[col];
            v2f b; b.x = f.x; b.y = f.y;
            acc[tIdx] = __builtin_amdgcn_wmma_f32_16x16x4_f32(false, a, false, b,
                                                              (short)0, acc[tIdx],
                                                              false, false);
        }
    }
    __syncthreads();   // end of sA/sH lifetime; gi/gh overlay begins

    float* out = is_h ? gh : gi;
    #pragma unroll
    for (int tIdx = 0; tIdx < 6; ++tIdx) {
        int col = (w4 * 6 + tIdx) * 16 + ncol;
        #pragma unroll
        for (int j = 0; j < 8; ++j) {
            int r = j + half * 8;
            out[r * GPAD + col] = acc[tIdx][j];
        }
    }
    __syncthreads();

    for (int i = threadIdx.x; i < 16 * DIM; i += 256) {
        int r = i >> 7;
        int d = i & 127;
        int grow = slab * 16 + r;
        float ir = gi[r * GPAD + d]       + bih[d];
        float iz = gi[r * GPAD + 128 + d] + bih[128 + d];
        float in = gi[r * GPAD + 256 + d] + bih[256 + d];
        float hr = gh[r * GPAD + d]       + bhh[d];
        float hz = gh[r * GPAD + 128 + d] + bhh[128 + d];
        float hn = gh[r * GPAD + 256 + d] + bhh[256 + d];
        float rg = 1.f / (1.f + __expf(-(ir + hr)));
        float zg = 1.f / (1.f + __expf(-(iz + hz)));
        float ng = tanhf(in + rg * hn);
        float ho = h[(size_t)grow * DIM + d];
        h[(size_t)grow * DIM + d] = (1.f - zg) * ng + zg * ho;
    }
}

// ---------------------------------------------------------------------------
__global__ void pool_kernel(const float* __restrict__ h,
                            const int* __restrict__ counts,
                            float* __restrict__ pooled) {
    int b = blockIdx.x;
    int d = threadIdx.x;   // 128 threads
    int off = 0;
    for (int i = 0; i < b; ++i) off += counts[i];
    int n = counts[b];
    float s = 0.f;
    for (int i = 0; i < n; ++i) s += h[(size_t)(off + i) * DIM + d];
    pooled[b * DIM + d] = s / (float)n;
}

__global__ void mlp_kernel(const float* __restrict__ pooled,
                           const float* __restrict__ W1,
                           const float* __restrict__ b1,
                           const float* __restrict__ W2,
                           const float* __restrict__ b2,
                           float* __restrict__ out) {
    __shared__ float partial[HID];
    int b = blockIdx.x;
    int j = threadIdx.x;  // 256 threads
    float s = b1[j];
    for (int d = 0; d < DIM; ++d) s += pooled[b * DIM + d] * W1[d * HID + j];
    s = fmaxf(s, 0.f);
    partial[j] = s * W2[j];
    __syncthreads();
    for (int st = 128; st > 0; st >>= 1) {
        if (j < st) partial[j] += partial[j + st];
        __syncthreads();
    }
    if (j == 0) out[b] = 1.f / (1.f + __expf(-(partial[0] + b2[0])));
}

// ---------------------------------------------------------------------------
extern "C" void kernel_launch(void* const* d_in, const int* in_sizes, int n_in,
                              void* d_out, int out_size, void* d_ws, size_t ws_size,
                              hipStream_t stream) {
    const float* node_features = (const float*)d_in[0];
    const int*   edge_src      = (const int*)d_in[1];
    const int*   edge_dst      = (const int*)d_in[2];
    const int*   edge_types    = (const int*)d_in[3];
    const int*   node_counts   = (const int*)d_in[4];
    const float* W_msg         = (const float*)d_in[5];
    const float* b_msg         = (const float*)d_in[6];
    const float* W_ih          = (const float*)d_in[7];
    const float* W_hh          = (const float*)d_in[8];
    const float* b_ih          = (const float*)d_in[9];
    const float* b_hh          = (const float*)d_in[10];
    const float* W1            = (const float*)d_in[11];
    const float* b1            = (const float*)d_in[12];
    const float* W2            = (const float*)d_in[13];
    const float* b2            = (const float*)d_in[14];

    char* ws = (char*)d_ws;
    size_t off = 0;
    auto alloc = [&](size_t bytes) {
        void* p = ws + off;
        off += (bytes + 255) & ~(size_t)255;
        return p;
    };
    float*  h      = (float*)alloc((size_t)N_NODES * DIM * sizeof(float));
    float*  Abuf   = (float*)alloc((size_t)N_NODES * DIM * sizeof(float));
    int*    cnt    = (int*)alloc((size_t)NKEYS * sizeof(int));
    int*    eoff   = (int*)alloc((size_t)NKEYS * sizeof(int));
    int*    cursor = (int*)alloc((size_t)NKEYS * sizeof(int));
    int*    eidx   = (int*)alloc((size_t)N_EDGES * sizeof(int));
    int*    bsum   = (int*)alloc(256 * sizeof(int));
    float*  pooled = (float*)alloc((size_t)B_GRAPHS * DIM * sizeof(float));
    float2* pkm    = (float2*)alloc((size_t)N_ETYPES * 32 * 2 * DIM * sizeof(float2));
    float2* pkih   = (float2*)alloc((size_t)32 * 2 * 384 * sizeof(float2));
    float2* pkhh   = (float2*)alloc((size_t)32 * 2 * 384 * sizeof(float2));
    (void)ws_size;

    // h <- node_features
    hipMemcpyAsync(h, node_features, (size_t)N_NODES * DIM * sizeof(float),
                   hipMemcpyDeviceToDevice, stream);

    // one-shot: packed weights
    pack_msg_kernel<<<(N_ETYPES * 32 * 2 * DIM + 255) / 256, 256, 0, stream>>>(W_msg, pkm);
    pack_gru_kernel<<<(32 * 2 * 384 + 255) / 256, 256, 0, stream>>>(W_ih, pkih);
    pack_gru_kernel<<<(32 * 2 * 384 + 255) / 256, 256, 0, stream>>>(W_hh, pkhh);

    // one-shot: CSR build (count -> exclusive scan -> fill)
    const int scanBlocks = (NKEYS + 1023) / 1024;   // 196
    hipMemsetAsync(cnt, 0, (size_t)NKEYS * sizeof(int), stream);
    count_edges_kernel<<<(N_EDGES + 255) / 256, 256, 0, stream>>>(edge_dst, edge_types, cnt);
    scan1_kernel<<<scanBlocks, 256, 0, stream>>>(cnt, eoff, bsum, NKEYS);
    scan2_kernel<<<1, 256, 0, stream>>>(bsum, scanBlocks);
    scan3_kernel<<<scanBlocks, 256, 0, stream>>>(eoff, bsum, NKEYS);
    hipMemcpyAsync(cursor, eoff, (size_t)NKEYS * sizeof(int),
                   hipMemcpyDeviceToDevice, stream);
    fill_csr_kernel<<<(N_EDGES + 255) / 256, 256, 0, stream>>>(edge_src, edge_dst,
                                                               edge_types, cursor, eidx);

    const int slabs = N_NODES / 16;   // 3125, exact
    for (int step = 0; step < N_STEPS; ++step) {
        msg_gemm_kernel<<<slabs, 128, 0, stream>>>(h, cnt, eoff, eidx, pkm, b_msg, Abuf);
        gru_fused_kernel<<<slabs, 256, 0, stream>>>(Abuf, h, pkih, pkhh, b_ih, b_hh);
    }

    pool_kernel<<<B_GRAPHS, DIM, 0, stream>>>(h, node_counts, pooled);
    mlp_kernel<<<B_GRAPHS, HID, 0, stream>>>(pooled, W1, b1, W2, b2, (float*)d_out);
}